// EMSLoss_5763846111461
// MI455X (gfx1250) — compile-verified
//
#include <hip/hip_runtime.h>
#include <hip/hip_bf16.h>
#include <cstdint>

// EMS loss: scale target logit by M, cross-entropy mean over batch.
// inputs: (4096, 32000) fp32, targets: (4096,) int32
#define N_CLASSES 32000
#define BATCH     4096
#define MARGIN    4.0f
#define THREADS   1024                  // 32 waves (wave32) -> fills a WGP
#define ROW_BYTES (N_CLASSES * 4)       // 128000 B, fits in 320 KB LDS
#define N4        (N_CLASSES / 4)       // 8000 float4 chunks = 7*1024 + 832
#define TAIL      (N4 - 7 * THREADS)    // 832 threads do an 8th chunk

__device__ __forceinline__ float waveReduceMax(float v) {
#pragma unroll
    for (int off = 16; off > 0; off >>= 1)
        v = fmaxf(v, __shfl_xor(v, off, 32));   // wave32!
    return v;
}

__device__ __forceinline__ float waveReduceSum(float v) {
#pragma unroll
    for (int off = 16; off > 0; off >>= 1)
        v += __shfl_xor(v, off, 32);            // wave32!
    return v;
}

// Async global->LDS copy, 16B/lane, non-temporal (data is read exactly once,
// 512 MB total > 192 MB L2, so NT avoids L2 thrash).
// INST_OFFSET is added to BOTH the LDS address and the global address (ISA 10.7),
// so one (lds, global) pair serves all 8 strided chunks via the immediate.
#define ASYNC_CP(OFF)                                                          \
    asm volatile("global_load_async_to_lds_b128 %0, %1, off offset:" #OFF      \
                 " th:TH_LOAD_NT"                                              \
                 :: "v"(lds_dst), "v"(src) : "memory")

__global__ __launch_bounds__(THREADS)
void ems_row_kernel(const float* __restrict__ inputs,
                    const int* __restrict__ targets,
                    float* __restrict__ row_nll) {
    extern __shared__ float row[];              // N_CLASSES floats (dynamic LDS)
    __shared__ float red[THREADS / 32];

    const int rowIdx = blockIdx.x;
    const int tid    = threadIdx.x;
    const int wave   = tid >> 5;
    const int lane   = tid & 31;
    const float* g   = inputs + (size_t)rowIdx * N_CLASSES;

    // ---- Stage the whole 128 KB row into LDS with async-to-LDS DMA ----
    // LDS byte address = low 32 bits of the generic shared pointer.
    const uint32_t lds_dst = (uint32_t)(uintptr_t)row + (uint32_t)tid * 16u;
    const float*   src     = g + tid * 4;
    ASYNC_CP(0);
    ASYNC_CP(16384);
    ASYNC_CP(32768);
    ASYNC_CP(49152);
    ASYNC_CP(65536);
    ASYNC_CP(81920);
    ASYNC_CP(98304);
    if (tid < TAIL) ASYNC_CP(114688);
    asm volatile("s_wait_asynccnt 0x0" ::: "memory");  // my wave's copies done
    __syncthreads();                                   // everyone's copies visible

    // ---- Apply margin to the target logit (in LDS) ----
    const int tgt = targets[rowIdx];
    if (tid == 0) row[tgt] *= MARGIN;
    __syncthreads();

    const float4* row4 = (const float4*)row;

    // ---- Pass 1: row max (128-bit LDS reads) ----
    float m = -INFINITY;
#pragma unroll
    for (int k = 0; k < 7; ++k) {
        float4 v = row4[tid + k * THREADS];
        m = fmaxf(m, fmaxf(fmaxf(v.x, v.y), fmaxf(v.z, v.w)));
    }
    if (tid < TAIL) {
        float4 v = row4[tid + 7 * THREADS];
        m = fmaxf(m, fmaxf(fmaxf(v.x, v.y), fmaxf(v.z, v.w)));
    }
    m = waveReduceMax(m);
    if (lane == 0) red[wave] = m;
    __syncthreads();
    if (wave == 0) {
        float v = red[lane];
        v = waveReduceMax(v);
        if (lane == 0) red[0] = v;
    }
    __syncthreads();
    const float rowMax = red[0];
    __syncthreads();   // done reading red before reuse

    // ---- Pass 2: sum of exp. Bare hardware v_exp_f32 is safe here:
    // arg = (x - max)*log2e <= 0, and underflow-to-zero is the desired
    // behavior for the sum. One FMA per element via the hoisted bias. ----
    const float LOG2E = 1.4426950408889634f;
    const float LN2   = 0.6931471805599453f;
    const float bias  = -rowMax * LOG2E;        // exp2(x*log2e + bias)
    float s = 0.0f;
#pragma unroll
    for (int k = 0; k < 7; ++k) {
        float4 v = row4[tid + k * THREADS];
        s += __builtin_amdgcn_exp2f(fmaf(v.x, LOG2E, bias));
        s += __builtin_amdgcn_exp2f(fmaf(v.y, LOG2E, bias));
        s += __builtin_amdgcn_exp2f(fmaf(v.z, LOG2E, bias));
        s += __builtin_amdgcn_exp2f(fmaf(v.w, LOG2E, bias));
    }
    if (tid < TAIL) {
        float4 v = row4[tid + 7 * THREADS];
        s += __builtin_amdgcn_exp2f(fmaf(v.x, LOG2E, bias));
        s += __builtin_amdgcn_exp2f(fmaf(v.y, LOG2E, bias));
        s += __builtin_amdgcn_exp2f(fmaf(v.z, LOG2E, bias));
        s += __builtin_amdgcn_exp2f(fmaf(v.w, LOG2E, bias));
    }
    s = waveReduceSum(s);
    if (lane == 0) red[wave] = s;
    __syncthreads();
    if (wave == 0) {
        float v = red[lane];
        v = waveReduceSum(v);
        if (lane == 0) {
            // sum in [1, 32000] -> bare v_log_f32 is safe.
            // logsumexp = max + ln(sum); nll = logsumexp - scaled_target_logit
            float lse = rowMax + __builtin_amdgcn_logf(v) * LN2;
            row_nll[rowIdx] = lse - row[tgt];
        }
    }
}

// Deterministic final mean over the 4096 per-row NLLs (no float atomics).
__global__ __launch_bounds__(1024)
void ems_final_reduce(const float* __restrict__ row_nll, float* __restrict__ out) {
    __shared__ float red[32];
    const int tid  = threadIdx.x;
    const int wave = tid >> 5;
    const int lane = tid & 31;
    float s = 0.0f;
    for (int i = tid; i < BATCH; i += 1024)
        s += row_nll[i];
    s = waveReduceSum(s);
    if (lane == 0) red[wave] = s;
    __syncthreads();
    if (wave == 0) {
        float v = red[lane];
        v = waveReduceSum(v);
        if (lane == 0) out[0] = v * (1.0f / (float)BATCH);
    }
}

extern "C" void kernel_launch(void* const* d_in, const int* in_sizes, int n_in,
                              void* d_out, int out_size, void* d_ws, size_t ws_size,
                              hipStream_t stream) {
    const float* inputs  = (const float*)d_in[0];
    const int*   targets = (const int*)d_in[1];
    float* row_nll = (float*)d_ws;      // 4096 floats of scratch
    float* out     = (float*)d_out;     // single float

    ems_row_kernel<<<BATCH, THREADS, ROW_BYTES, stream>>>(inputs, targets, row_nll);
    ems_final_reduce<<<1, 1024, 0, stream>>>(row_nll, out);
}